// ViterbiDecoder_90640989815180
// MI455X (gfx1250) — compile-verified
//
#include <hip/hip_runtime.h>

// Viterbi decode (max-plus forward + parallel backtrace), B=128, T=1024, N=24.
// One wave32 per batch element. phi tiles streamed into an LDS ring buffer with
// gfx1250 async-to-LDS loads (ASYNCcnt-tracked), depth-8 prefetch.
// alpha broadcast via LDS (6 x ds_load_b128) instead of 24 readlanes.
// Backtrace parallelized as 32 blocks x 32 steps (composed backpointer maps).

#define T_LEN   1024
#define NTAG    24
#define STEPF   (NTAG * NTAG)      // 576 floats per time step
#define STEPB   (STEPF * 4)        // 2304 bytes per time step
#define RING    8                  // prefetch depth (slots in LDS ring)
#define BPS     25                 // padded backpointer row stride (bytes)

// 4 x b128 covers bytes [0,2048), 1 x b64 covers [2048,2304).
#define ASYNC_B128(ldsa, voff, sbase, OFF)                                   \
  asm volatile("global_load_async_to_lds_b128 %0, %1, %2 offset:" OFF        \
               :: "v"(ldsa), "v"(voff), "s"(sbase) : "memory")
#define ASYNC_B64(ldsa, voff, sbase, OFF)                                    \
  asm volatile("global_load_async_to_lds_b64 %0, %1, %2 offset:" OFF         \
               :: "v"(ldsa), "v"(voff), "s"(sbase) : "memory")
#define WAIT_ASYNC(NSTR)                                                     \
  asm volatile("s_wait_asynccnt " NSTR ::: "memory")
#define WAIT_DS0()                                                           \
  asm volatile("s_wait_dscnt 0x0" ::: "memory")

__global__ __launch_bounds__(32)
void viterbi_kernel(const float* __restrict__ lp,
                    const unsigned char* __restrict__ mask,
                    int* __restrict__ out)
{
    __shared__ float          ring[RING * STEPF];        // 18432 B, DMA-written only
    __shared__ unsigned char  bp[(T_LEN + 1) * BPS];     // 25625 B (row 1024 = identity)
    __shared__ __attribute__((aligned(16))) float alphaLDS[NTAG];
    __shared__ unsigned char  Gmap[32 * NTAG];           // composed block maps
    __shared__ unsigned char  btagS[33];                 // block-boundary tags
    __shared__ uint4          maskb4[T_LEN / 16];        // 1024 B of mask bytes
    __shared__ unsigned char  tagsb[T_LEN];

    const int lane = (int)threadIdx.x;
    const int b    = (int)blockIdx.x;
    const int jj   = lane < NTAG ? lane : (NTAG - 1);    // lanes 24..31 shadow j=23

    const unsigned long long sbase =
        (unsigned long long)(lp + (size_t)b * T_LEN * STEPF);
    const unsigned ringBase = (unsigned)(uintptr_t)(&ring[0]);  // LDS byte offset

    const unsigned gl128 = (unsigned)lane * 16u;
    const unsigned gl64  = 2048u + (unsigned)lane * 8u;

    // ---- prologue: prefetch steps 0..RING-1 into slots 0..RING-1 ----
    #pragma unroll
    for (int p = 0; p < RING; ++p) {
        unsigned g128 = (unsigned)(p * STEPB) + gl128;
        unsigned g64  = (unsigned)(p * STEPB) + gl64;
        unsigned l128 = ringBase + (unsigned)(p * STEPB) + gl128;
        unsigned l64  = ringBase + (unsigned)(p * STEPB) + gl64;
        ASYNC_B128(l128, g128, sbase, "0");
        ASYNC_B128(l128, g128, sbase, "512");
        ASYNC_B128(l128, g128, sbase, "1024");
        ASYNC_B128(l128, g128, sbase, "1536");
        ASYNC_B64 (l64,  g64,  sbase, "0");
    }

    // mask bytes to LDS early: HBM latency hides behind the first prefetch wait
    {
        const uint4* mrow = (const uint4*)(mask + (size_t)b * T_LEN);
        maskb4[lane]      = mrow[lane];
        maskb4[lane + 32] = mrow[lane + 32];
    }

    // identity map in virtual row t=1024 (used by block 31 of the backtrace)
    if (lane < NTAG) bp[T_LEN * BPS + lane] = (unsigned char)lane;

    // ---- t = 0 : alpha0[j] = max_i phi[0][i][j] ----
    WAIT_ASYNC("35");                         // (RING-1)*5 outstanding => step 0 landed
    {
        float a0 = -3.4e38f, a1 = -3.4e38f;
        #pragma unroll
        for (int k = 0; k < NTAG / 2; ++k) {
            a0 = fmaxf(a0, ring[k * NTAG + jj]);
            a1 = fmaxf(a1, ring[(k + 12) * NTAG + jj]);
        }
        if (lane < NTAG) alphaLDS[lane] = fmaxf(a0, a1);
    }
    WAIT_DS0();                               // slot-0 reads done before refill
    {   // prefetch step RING into slot 0
        int p = RING;
        unsigned g128 = (unsigned)(p * STEPB) + gl128;
        unsigned g64  = (unsigned)(p * STEPB) + gl64;
        unsigned l128 = ringBase + gl128;
        unsigned l64  = ringBase + gl64;
        ASYNC_B128(l128, g128, sbase, "0");
        ASYNC_B128(l128, g128, sbase, "512");
        ASYNC_B128(l128, g128, sbase, "1024");
        ASYNC_B128(l128, g128, sbase, "1536");
        ASYNC_B64 (l64,  g64,  sbase, "0");
    }

    int slot = 1;
    // ---- main steps with steady-state prefetch: t = 1 .. T-RING-1 ----
    for (int t = 1; t <= T_LEN - RING - 1; ++t) {
        WAIT_ASYNC("35");                     // step t's 5 async ops complete
        const int sb = slot * STEPF;

        float aa[NTAG];                       // broadcast of alpha[t-1][0..23]
        #pragma unroll
        for (int q = 0; q < NTAG / 4; ++q) {
            float4 v = *(const float4*)(&alphaLDS[4 * q]);
            aa[4 * q + 0] = v.x; aa[4 * q + 1] = v.y;
            aa[4 * q + 2] = v.z; aa[4 * q + 3] = v.w;
        }

        float b0 = -3.4e38f, b1 = -3.4e38f;
        int   i0 = 0,        i1 = 12;
        #pragma unroll
        for (int k = 0; k < NTAG / 2; ++k) {
            float v0 = aa[k]      + ring[sb + k * NTAG + jj];
            float v1 = aa[k + 12] + ring[sb + (k + 12) * NTAG + jj];
            if (v0 > b0) { b0 = v0; i0 = k; }        // strict > : first-max
            if (v1 > b1) { b1 = v1; i1 = k + 12; }
        }
        float best = b0; int bi = i0;
        if (b1 > b0) { best = b1; bi = i1; }         // ties -> lower half (smaller i)

        if (lane < NTAG) {
            alphaLDS[lane]      = best;
            bp[t * BPS + lane]  = (unsigned char)bi;
        }

        WAIT_DS0();                    // this slot's reads complete before DMA refill
        {
            int p = t + RING;          // <= T-1 in this loop
            unsigned g128 = (unsigned)(p * STEPB) + gl128;
            unsigned g64  = (unsigned)(p * STEPB) + gl64;
            unsigned l128 = ringBase + (unsigned)(slot * STEPB) + gl128;
            unsigned l64  = ringBase + (unsigned)(slot * STEPB) + gl64;
            ASYNC_B128(l128, g128, sbase, "0");
            ASYNC_B128(l128, g128, sbase, "512");
            ASYNC_B128(l128, g128, sbase, "1024");
            ASYNC_B128(l128, g128, sbase, "1536");
            ASYNC_B64 (l64,  g64,  sbase, "0");
        }
        slot = (slot + 1 == RING) ? 0 : slot + 1;
    }

    // ---- tail steps: t = T-RING .. T-1 (drain; no more prefetch) ----
    for (int t = T_LEN - RING; t < T_LEN; ++t) {
        WAIT_ASYNC("0");
        const int sb = slot * STEPF;

        float aa[NTAG];
        #pragma unroll
        for (int q = 0; q < NTAG / 4; ++q) {
            float4 v = *(const float4*)(&alphaLDS[4 * q]);
            aa[4 * q + 0] = v.x; aa[4 * q + 1] = v.y;
            aa[4 * q + 2] = v.z; aa[4 * q + 3] = v.w;
        }

        float b0 = -3.4e38f, b1 = -3.4e38f;
        int   i0 = 0,        i1 = 12;
        #pragma unroll
        for (int k = 0; k < NTAG / 2; ++k) {
            float v0 = aa[k]      + ring[sb + k * NTAG + jj];
            float v1 = aa[k + 12] + ring[sb + (k + 12) * NTAG + jj];
            if (v0 > b0) { b0 = v0; i0 = k; }
            if (v1 > b1) { b1 = v1; i1 = k + 12; }
        }
        float best = b0; int bi = i0;
        if (b1 > b0) { best = b1; bi = i1; }

        if (lane < NTAG) {
            alphaLDS[lane]     = best;
            bp[t * BPS + lane] = (unsigned char)bi;
        }
        slot = (slot + 1 == RING) ? 0 : slot + 1;
    }
    __syncthreads();

    // ---- Phase A: compose 32 block maps (32 steps each), chains interleaved ----
    // Block g composes f_t for t = 32g+32 .. 32g+1 (t=1024 is the identity row),
    // mapping tag@time(32g+32) -> tag@time(32g). Lane j chases tag j.
    // Rolled k-loop: each round issues 32 independent ds_load_u8 (latency hidden),
    // keeping static code small and waits batched.
    if (lane < NTAG) {
        unsigned xg[32];
        #pragma unroll
        for (int g = 0; g < 32; ++g) xg[g] = (unsigned)lane;
        #pragma unroll 1
        for (int k = 32; k >= 1; --k) {
            #pragma unroll
            for (int g = 0; g < 32; ++g) {
                xg[g] = bp[(unsigned)(32 * g + k) * BPS + xg[g]];
            }
        }
        #pragma unroll
        for (int g = 0; g < 32; ++g)
            Gmap[g * NTAG + lane] = (unsigned char)xg[g];
    }
    __syncthreads();

    // ---- Phase B: serial walk over 32 composed maps (lane 0) ----
    if (lane == 0) {
        int last = 0; float bv = alphaLDS[0];
        #pragma unroll
        for (int j = 1; j < NTAG; ++j) {
            float v = alphaLDS[j];
            if (v > bv) { bv = v; last = j; }         // first-max like jnp.argmax
        }
        unsigned x = (unsigned)last;                   // tag @ virtual time 1024
        btagS[32] = (unsigned char)x;
        #pragma unroll 1
        for (int g = 31; g >= 0; --g) {
            x = Gmap[g * NTAG + x];                    // tag @ time 32g
            btagS[g] = (unsigned char)x;
        }
    }
    __syncthreads();

    // ---- Phase C: in-block walks, one block per lane ----
    {
        unsigned x = btagS[lane + 1];                  // tag @ time 32*(lane+1)
        #pragma unroll 1
        for (int k = 32; k >= 1; --k) {
            x = bp[(unsigned)(32 * lane + k) * BPS + x];
            tagsb[32 * lane + k - 1] = (unsigned char)x;
        }
    }
    __syncthreads();

    // ---- masked, coalesced int32 output ----
    const unsigned char* maskb = (const unsigned char*)maskb4;
    int* orow = out + (size_t)b * T_LEN;
    #pragma unroll 4
    for (int k = 0; k < T_LEN / 32; ++k) {
        int idx = k * 32 + lane;
        orow[idx] = maskb[idx] ? (int)tagsb[idx] : -1;
    }
}

extern "C" void kernel_launch(void* const* d_in, const int* in_sizes, int n_in,
                              void* d_out, int out_size, void* d_ws, size_t ws_size,
                              hipStream_t stream)
{
    const float*         lp   = (const float*)d_in[0];
    const unsigned char* mask = (const unsigned char*)d_in[1];   // jax bool -> 1 byte
    int*                 out  = (int*)d_out;                     // int32 tags / -1

    const int B = in_sizes[0] / (T_LEN * STEPF);                 // = 128
    viterbi_kernel<<<dim3(B), dim3(32), 0, stream>>>(lp, mask, out);
}